// ProjectLayer_23167053594904
// MI455X (gfx1250) — compile-verified
//
#include <hip/hip_runtime.h>
#include <stdint.h>

#define B_ROWS   1024
#define M_COLS   128
#define H_TOTAL  672
#define NPART    3
#define MBUCK    0x7FFFFFFFu   // 2^31 - 1 (Mersenne prime)
#define MHALF    0x3FFFFFFFu   // MBUCK >> 1

// Centered hash: ((s*w) mod (2^31-1)) mapped to [-HALF, HALF], as float.
// s,w < 2^31 so the product < 2^62; two folds using 2^31 == 1 (mod M) leave
// u <= 2^31, and a single select finishes BOTH the mod and the centering:
//   u <= HALF      -> u        (positive half)
//   HALF < u < M   -> u - M    (negative half)
//   u == M         -> 0        (residue 0)
//   u == M+1=2^31  -> 1        (residue 1)
__device__ __forceinline__ float hash_centered(unsigned s, unsigned w) {
  const unsigned long long pr = (unsigned long long)s * w;     // v_mad_u64_u32
  const unsigned t = (unsigned)(pr >> 31) + ((unsigned)pr & MBUCK);  // < 2^32
  const unsigned u = (t >> 31) + (t & MBUCK);                        // <= 2^31
  const int c = (int)(u - ((u > MHALF) ? MBUCK : 0u));
  return (float)c;
}

// One block per batch row b. 22 waves (704 threads):
//   waves 0..3  -> partition 0, h = wave*32+lane         (valid h < 112)
//   waves 4..10 -> partition 1, h = 112+(wave-4)*32+lane  (exactly 224)
//   waves 11..21-> partition 2, h = 336+(wave-11)*32+lane (valid h < 672)
// Each wave is fully inside one partition => wave-uniform sig row + trip count.
__global__ __launch_bounds__(704) void sgnn_project_kernel(
    const long long* __restrict__ sig,       // [3,1024,128] int64 in [0, 2^31)
    const unsigned char* __restrict__ mask,  // [3,1024,128] bool (1 byte each)
    const long long* __restrict__ seeds,     // [672] int64 in [1, 2^31)
    float* __restrict__ out)                 // [1024,672] f32
{
  __shared__ unsigned int sRaw[NPART * M_COLS];                 // staged sig rows (low 32b)
  __shared__ __align__(16) unsigned int sSig[NPART * M_COLS];   // mask-compacted sig rows
  __shared__ int   sCnt[NPART];
  __shared__ float sInv[NPART];                                 // 1 / (count * HALF)

  const int b    = blockIdx.x;
  const int tid  = threadIdx.x;
  const int wave = tid >> 5;
  const int lane = tid & 31;

  // ---- Phase 1: async global->LDS stage of the 3 sig rows for this b ------
  // int64 little-endian and value < 2^31: the low dword IS the value.
  if (tid < NPART * M_COLS) {              // waves 0..11, full waves only
    const int p = tid >> 7;
    const int m = tid & (M_COLS - 1);
    const unsigned long long gaddr =
        (unsigned long long)(uintptr_t)(sig + ((size_t)p * B_ROWS + b) * M_COLS + m);
    const unsigned int ldsoff = (unsigned int)(uintptr_t)&sRaw[tid]; // low 32b of flat = LDS offset
    asm volatile("global_load_async_to_lds_b32 %0, %1, off"
                 :: "v"(ldsoff), "v"(gaddr) : "memory");
  }
  asm volatile("s_wait_asynccnt 0x0" ::: "memory");
  __syncthreads();

  // ---- Phase 2: per-partition mask compaction (wave p handles partition p) --
  if (wave < NPART) {
    const int p = wave;
    const unsigned char* mrow = mask + ((size_t)p * B_ROWS + b) * M_COLS;
    unsigned base = 0;
    for (int c = 0; c < M_COLS / 32; ++c) {
      const int m = c * 32 + lane;
      const bool v = mrow[m] != 0;
      const unsigned bal = __builtin_amdgcn_ballot_w32(v);
      const unsigned pre = (unsigned)__builtin_popcount(bal & ((1u << lane) - 1u));
      const unsigned sv  = sRaw[p * M_COLS + m];
      if (v) sSig[p * M_COLS + base + pre] = sv;
      base += (unsigned)__builtin_popcount(bal);
    }
    if (lane == 0) {
      sCnt[p] = (int)base;
      const unsigned d = base ? base : 1u;                  // ragged max(count,1)
      sInv[p] = 1.0f / ((float)d * 1073741823.0f);          // mean and /HALF fused
    }
  }
  __syncthreads();

  // ---- Phase 3: hash-project. lane = h, sig values wave-uniform ------------
  int p, h;
  if (wave < 4)       { p = 0; h = wave * 32 + lane; }
  else if (wave < 11) { p = 1; h = 112 + (wave - 4) * 32 + lane; }
  else                { p = 2; h = 336 + (wave - 11) * 32 + lane; }
  const int  hEnd   = (p == 0) ? 112 : (p == 1 ? 336 : 672);
  const bool active = h < hEnd;

  const unsigned w  = active ? (unsigned)seeds[h] : 0u;     // seed < 2^31 fits u32
  const int cnt = __builtin_amdgcn_readfirstlane(sCnt[p]);  // scalar loop control
  const unsigned* __restrict__ row = &sSig[p * M_COLS];

  // Four independent accumulators (breaks the f32-add chain, enables VOPD),
  // ds_load_b128 per 4 elements (16B-aligned since quad index is multiple of 4).
  float a0 = 0.0f, a1 = 0.0f, a2 = 0.0f, a3 = 0.0f;
  const int cnt4 = cnt & ~3;
  int i = 0;
  #pragma unroll 2
  for (; i < cnt4; i += 4) {
    const uint4 s4 = *(const uint4*)(row + i);              // ds_load_b128 broadcast
    a0 += hash_centered(s4.x, w);
    a1 += hash_centered(s4.y, w);
    a2 += hash_centered(s4.z, w);
    a3 += hash_centered(s4.w, w);
  }
  for (; i < cnt; ++i)                                      // tail (<= 3 elements)
    a0 += hash_centered(row[i], w);

  const float acc = (a0 + a1) + (a2 + a3);
  if (active) out[(size_t)b * H_TOTAL + h] = acc * sInv[p];
}

extern "C" void kernel_launch(void* const* d_in, const int* in_sizes, int n_in,
                              void* d_out, int out_size, void* d_ws, size_t ws_size,
                              hipStream_t stream) {
  (void)in_sizes; (void)n_in; (void)out_size; (void)d_ws; (void)ws_size;
  const long long*     sig   = (const long long*)d_in[0];      // signatures [3,1024,128] int64
  const unsigned char* mask  = (const unsigned char*)d_in[1];  // mask [3,1024,128] bool (1B)
  const long long*     seeds = (const long long*)d_in[2];      // hash_seed [672] int64
  float*               out   = (float*)d_out;                  // [1024,672] f32

  dim3 grid(B_ROWS), block(704);
  sgnn_project_kernel<<<grid, block, 0, stream>>>(sig, mask, seeds, out);
}